// RoIPooling_63513976373634
// MI455X (gfx1250) — compile-verified
//
#include <hip/hip_runtime.h>
#include <hip/hip_bf16.h>

// RoIPool for MI455X (gfx1250).
// One 256-thread block per (roi, 4-channel chunk). A single 3D Tensor Data
// Mover descriptor DMAs the RoI's rectangular tile for 4 consecutive channel
// planes into LDS (tile_dim2=4 walks the channel axis via tensor_dim1_stride
// = 2500). Wave 0 issues the TDM, waits on TENSORcnt, workgroup barrier, then
// 196 lanes compute the 4x7x7 max-pool bins out of LDS. Whole feature map
// (5.1MB) is L2-resident; latency of each DMA is hidden by ~8 co-resident
// blocks per WGP (40KB worst-case LDS each of 320KB).

#define OUT_HW 7
#define FH_ 50
#define FW_ 50
#define C_ 256
#define CH_ 4   // channels per block / per TDM descriptor

typedef __attribute__((ext_vector_type(4))) unsigned int v4u_;
typedef __attribute__((ext_vector_type(8))) int v8i_;
typedef __attribute__((ext_vector_type(4))) int v4i_;

__global__ void roipool_tdm_kernel(const float* __restrict__ feat,
                                   const float* __restrict__ rois,
                                   const int* __restrict__ img_h_p,
                                   const int* __restrict__ img_w_p,
                                   float* __restrict__ out) {
    extern __shared__ float tile[];  // CH_ * 50*50 floats = 40000 B max

    const int blk = blockIdx.x;
    const int chunks = C_ / CH_;                 // 64
    const int n = blk / chunks;
    const int c_base = (blk % chunks) * CH_;

    const float sh = (float)FH_ / (float)img_h_p[0];
    const float sw = (float)FW_ / (float)img_w_p[0];

    const float* r = rois + (size_t)n * 5;
    const int bidx = (int)r[0];
    int x1 = max((int)floorf(r[1] * sw), 0);
    int y1 = max((int)floorf(r[2] * sh), 0);
    int x2 = min((int)floorf(r[3] * sw), FW_);
    int y2 = min((int)floorf(r[4] * sh), FH_);
    const int roi_w = max(x2 - x1, 1);
    const int roi_h = max(y2 - y1, 1);

    // Bounding tile of every bin's (clipped) pooling window.
    const int col0 = min(x1, FW_ - 1);
    const int row0 = min(y1, FH_ - 1);
    const int tw = max(min(x1 + roi_w, FW_) - col0, 1);
    const int th = max(min(y1 + roi_h, FH_) - row0, 1);

    const unsigned long long gaddr =
        (unsigned long long)(const void*)(feat +
            ((size_t)(bidx * C_ + c_base) * (FH_ * FW_) +
             (size_t)row0 * FW_ + col0));

#if __has_builtin(__builtin_amdgcn_tensor_load_to_lds)
    if (threadIdx.x < 32) {  // wave 0 issues the TDM op (EXEC-safe either way)
        v4u_ g0;
        v8i_ g1;
        v4i_ g2;
        v4i_ g3 = {0, 0, 0, 0};
        // ---- D# group 0 (128b): count=1 valid user desc, lds_addr=0 (dyn LDS
        // base), 57-bit global byte address, type=2 ("image").
        g0.x = 1u;
        g0.y = 0u;
        g0.z = (unsigned int)(gaddr & 0xFFFFFFFFu);
        g0.w = (unsigned int)((gaddr >> 32) & 0x01FFFFFFu) | (2u << 30);
        // ---- D# group 1 (256b): no multicast, data_size=4B (code 2),
        // tensor_dim0/1 = 50x50, tile = tw x th x CH_,
        // dim0 stride = 50 (row), dim1 stride = 2500 (channel plane).
        const unsigned int td0 = FW_, td1 = FH_;
        const unsigned int st0 = FW_, st1 = FH_ * FW_;
        g1.s0 = (int)(2u << 16);                                   // data_size=4B
        g1.s1 = (int)((td0 & 0xFFFFu) << 16);                      // tdim0 lo16
        g1.s2 = (int)(((td0 >> 16) & 0xFFFFu) | ((td1 & 0xFFFFu) << 16));
        g1.s3 = (int)(((td1 >> 16) & 0xFFFFu) | (((unsigned)tw & 0xFFFFu) << 16));
        g1.s4 = (int)(((unsigned)th & 0xFFFFu) | ((unsigned)CH_ << 16)); // tile1|tile2
        g1.s5 = (int)st0;                                          // stride0 lo32
        g1.s6 = (int)((st1 & 0xFFFFu) << 16);                      // stride0 hi=0 | stride1 lo16
        g1.s7 = (int)((st1 >> 16) & 0xFFFFFFFFu);                  // stride1 [47:16]
        // ---- D# group 2 (128b): tensor_dim2 = C (OOB bound on channel walk),
        // tensor_dim3 / stride2 / tile_dim3 unused (= 0).
        g2.x = C_;
        g2.y = 0;
        g2.z = 0;
        g2.w = 0;
#if defined(__clang_major__) && __clang_major__ >= 23
        v8i_ g4 = {0, 0, 0, 0, 0, 0, 0, 0};
        __builtin_amdgcn_tensor_load_to_lds(g0, g1, g2, g3, g4, 0);
#else
        __builtin_amdgcn_tensor_load_to_lds(g0, g1, g2, g3, 0);
#endif
        __builtin_amdgcn_s_wait_tensorcnt(0);
    }
#else
    // Fallback: cooperative strided copy into LDS (same packed layout).
    {
        const float* plane = (const float*)(const void*)gaddr;
        const int per_ch = tw * th;
        for (int i = threadIdx.x; i < per_ch * CH_; i += blockDim.x) {
            int ch = i / per_ch;
            int rem = i - ch * per_ch;
            int yy = rem / tw, xx = rem - yy * tw;
            tile[i] = plane[(size_t)ch * (FH_ * FW_) + (size_t)yy * FW_ + xx];
        }
    }
#endif
    __syncthreads();

    const int t = threadIdx.x;
    if (t < CH_ * OUT_HW * OUT_HW) {
        const int ch = t / (OUT_HW * OUT_HW);
        const int bin = t - ch * (OUT_HW * OUT_HW);
        const int oh = bin / OUT_HW;
        const int ow = bin - oh * OUT_HW;

        const float bin_h = (float)roi_h / (float)OUT_HW;
        const float bin_w = (float)roi_w / (float)OUT_HW;
        int ys = min(max((int)floorf((float)y1 + (float)oh * bin_h), 0), FH_ - 1);
        int ye = min(max((int)floorf((float)y1 + (float)(oh + 1) * bin_h), 0), FH_);
        int xs = min(max((int)floorf((float)x1 + (float)ow * bin_w), 0), FW_ - 1);
        int xe = min(max((int)floorf((float)x1 + (float)(ow + 1) * bin_w), 0), FW_);

        float v = 0.0f;  // empty bins stay 0 (reference semantics)
        if ((ye > ys) && (xe > xs)) {
            const float* chtile = tile + (size_t)ch * tw * th;
            float m = -__builtin_inff();
            for (int y = ys; y < ye; ++y) {
                const float* rowp = chtile + (size_t)(y - row0) * tw - col0;
                for (int x = xs; x < xe; ++x) m = fmaxf(m, rowp[x]);
            }
            v = m;
        }
        out[((size_t)n * C_ + (c_base + ch)) * (OUT_HW * OUT_HW) + bin] = v;
    }
}

extern "C" void kernel_launch(void* const* d_in, const int* in_sizes, int n_in,
                              void* d_out, int out_size, void* d_ws, size_t ws_size,
                              hipStream_t stream) {
    const float* feat = (const float*)d_in[0];
    const float* rois = (const float*)d_in[1];
    const int* img_h = (const int*)d_in[2];
    const int* img_w = (const int*)d_in[3];
    float* out = (float*)d_out;

    const int N = in_sizes[1] / 5;  // rois are [N,5]
    dim3 grid((unsigned)(N * (C_ / CH_)));
    dim3 block(256);
    size_t shmem = (size_t)CH_ * FH_ * FW_ * sizeof(float);  // 40000 B
    hipLaunchKernelGGL(roipool_tdm_kernel, grid, block, shmem, stream,
                       feat, rois, img_h, img_w, out);
}